// GINEncoder_83038897701199
// MI455X (gfx1250) — compile-verified
//
#include <hip/hip_runtime.h>

typedef __attribute__((ext_vector_type(2))) float v2f;
typedef __attribute__((ext_vector_type(8))) float v8f;

#define GDIM 128

// ---------------------------------------------------------------- zero buffer
__global__ __launch_bounds__(256) void gin_zero(float4* p, long n4) {
  long i = (long)blockIdx.x * blockDim.x + threadIdx.x;
  long stride = (long)gridDim.x * blockDim.x;
  float4 z = {0.f, 0.f, 0.f, 0.f};
  for (; i < n4; i += stride) p[i] = z;
}

// ------------------------------------------------- edge scatter (segment sum)
// One wave32 per edge: lane l handles floats [4l, 4l+4) of the 128-dim row.
__global__ __launch_bounds__(256) void gin_scatter(
    const float* __restrict__ x, const int* __restrict__ src,
    const int* __restrict__ dst, float* agg, int nedges) {
  int w = (int)((blockIdx.x * 256u + threadIdx.x) >> 5);
  int lane = threadIdx.x & 31;
  if (w >= nedges) return;
  int s = src[w];
  int d = dst[w];
  const float4 v = *(const float4*)(x + (long)s * GDIM + lane * 4);
  float* o = agg + (long)d * GDIM + lane * 4;
  atomicAdd(o + 0, v.x);
  atomicAdd(o + 1, v.y);
  atomicAdd(o + 2, v.z);
  atomicAdd(o + 3, v.w);
}

// ---------------------------------------------------------- fp32 WMMA GEMM
// Out[n,128] = act( (A0 [+ A1]) @ W + bias ),  W is 128x128 row-major.
// 8 waves/block; each wave computes a 32x128 output tile (two 16-row M-tiles
// sharing every B fragment) with v_wmma_f32_16x16x4_f32:
// per K-step = 8 B ds_load_b64  ->  16 WMMA  (2 WMMA per LDS load).
//
// LDS W layout is pair-interleaved: dword idx = p*256 + 2n holds
// (W[2p][n], W[2p+1][n]) so every B fragment is one aligned ds_load_b64.
template <bool HASADD, bool RELU>
__global__ __launch_bounds__(256) void gin_gemm(
    const float* A0, const float* A1, const float* __restrict__ W,
    const float* __restrict__ bias, float* Out, int nrows) {
  __shared__ float sW[GDIM * GDIM];  // 64 KB of the WGP's 320 KB LDS
  int tid = threadIdx.x;
  for (int i = tid; i < (GDIM / 2) * GDIM; i += 256) {
    int p = i >> 7;   // row pair 0..63
    int n = i & 127;  // column
    int idx = p * 256 + 2 * n;
    sW[idx]     = W[(2 * p) * GDIM + n];
    sW[idx + 1] = W[(2 * p + 1) * GDIM + n];
  }
  __syncthreads();

  int wave = tid >> 5;
  int lane = tid & 31;
  long mbase = ((long)blockIdx.x * 8 + wave) * 32;
  if (mbase >= nrows) return;          // wave-uniform: EXEC all-ones at WMMA
  bool has2 = (mbase + 16) < nrows;    // second 16-row tile valid?
  long mbase2 = has2 ? (mbase + 16) : mbase;

  int lmod = lane & 15;
  int par = lane >> 4;  // 0: K=k,k+1   1: K=k+2,k+3
  int khalf = par << 1;
  int mhi = par << 3;   // C/D layout: VGPR j holds rows j (lo) / j+8 (hi)

  const float* arA0 = A0 + (mbase + lmod) * GDIM;
  const float* arB0 = A0 + (mbase2 + lmod) * GDIM;
  const float* arA1 = HASADD ? (A1 + (mbase + lmod) * GDIM) : nullptr;
  const float* arB1 = HASADD ? (A1 + (mbase2 + lmod) * GDIM) : nullptr;
  const float* sWp = sW + par * 256 + 2 * lmod;  // + (k>>1)*256 per K-step

  // A prefetch pipeline (one K-step ahead), both tiles
  v2f pa0 = *(const v2f*)(arA0 + khalf);
  v2f pb0 = *(const v2f*)(arB0 + khalf);
  v2f pa1, pb1;
  if (HASADD) {
    pa1 = *(const v2f*)(arA1 + khalf);
    pb1 = *(const v2f*)(arB1 + khalf);
  }

  v8f acc[16] = {};  // [nt] = tile0, [8+nt] = tile1

  for (int k = 0; k < GDIM; k += 4) {
    v2f aA = pa0, aB = pb0;
    if (HASADD) {
      aA.x += pa1.x; aA.y += pa1.y;
      aB.x += pb1.x; aB.y += pb1.y;
    }
    if (k + 4 < GDIM) {
      pa0 = *(const v2f*)(arA0 + (k + 4) + khalf);
      pb0 = *(const v2f*)(arB0 + (k + 4) + khalf);
      if (HASADD) {
        pa1 = *(const v2f*)(arA1 + (k + 4) + khalf);
        pb1 = *(const v2f*)(arB1 + (k + 4) + khalf);
      }
    }
    const float* wp = sWp + (k >> 1) * 256;
    v2f b[8];
#pragma unroll
    for (int nt = 0; nt < 8; nt++)
      b[nt] = *(const v2f*)(wp + nt * 32);  // ds_load_b64, immediate offset
#pragma unroll
    for (int nt = 0; nt < 8; nt++) {
      acc[nt] = __builtin_amdgcn_wmma_f32_16x16x4_f32(
          false, aA, false, b[nt], (short)0, acc[nt], false, false);
      acc[8 + nt] = __builtin_amdgcn_wmma_f32_16x16x4_f32(
          false, aB, false, b[nt], (short)0, acc[8 + nt], false, false);
    }
  }

#pragma unroll
  for (int nt = 0; nt < 8; nt++) {
    int ncol = nt * 16 + lmod;
    float bv = bias[ncol];
#pragma unroll
    for (int j = 0; j < 8; j++) {
      float v = acc[nt][j] + bv;
      if (RELU) v = fmaxf(v, 0.f);
      Out[(mbase + mhi + j) * GDIM + ncol] = v;
    }
    if (has2) {
#pragma unroll
      for (int j = 0; j < 8; j++) {
        float v = acc[8 + nt][j] + bv;
        if (RELU) v = fmaxf(v, 0.f);
        Out[(mbase2 + mhi + j) * GDIM + ncol] = v;
      }
    }
  }
}

// ---------------------------------------------------------------- launcher
extern "C" void kernel_launch(void* const* d_in, const int* in_sizes, int n_in,
                              void* d_out, int out_size, void* d_ws,
                              size_t ws_size, hipStream_t stream) {
  const float* x_in = (const float*)d_in[0];
  const int* ei = (const int*)d_in[1];  // [2, NE]: row0 = src, row1 = dst
  const float* W1 = (const float*)d_in[2];
  const float* b1 = (const float*)d_in[3];
  const float* W2 = (const float*)d_in[4];
  const float* b2 = (const float*)d_in[5];
  float* out = (float*)d_out;

  const int N = in_sizes[0] / GDIM;
  const int NE = in_sizes[1] / 2;
  const long nd = (long)N * GDIM;

  float* aggh = (float*)d_ws;  // agg, then reused (per-row safely) as h
  float* x1 = aggh + nd;       // layer-0 output

  const int mtiles32 = (N + 31) / 32;          // 32 rows per wave
  const int gemm_blocks = (mtiles32 + 7) / 8;  // 8 waves per block
  const int scat_blocks = (NE + 7) / 8;
  const int zero_blocks = (int)((nd / 4 + 255) / 256);

  const float* xc = x_in;
  for (int layer = 0; layer < 3; ++layer) {
    float* xn = (layer == 0) ? x1 : out;  // L1 -> out, L2 reads out, rewrites out
    gin_zero<<<zero_blocks, 256, 0, stream>>>((float4*)aggh, nd / 4);
    gin_scatter<<<scat_blocks, 256, 0, stream>>>(xc, ei, ei + NE, aggh, NE);
    gin_gemm<true, true><<<gemm_blocks, 256, 0, stream>>>(
        xc, aggh, W1 + (long)layer * GDIM * GDIM, b1 + layer * GDIM, aggh, N);
    gin_gemm<false, false><<<gemm_blocks, 256, 0, stream>>>(
        aggh, nullptr, W2 + (long)layer * GDIM * GDIM, b2 + layer * GDIM, xn, N);
    xc = xn;
  }
}